// RIME_75239237091345
// MI455X (gfx1250) — compile-verified
//
#include <hip/hip_runtime.h>
#include <math.h>

typedef float v2f __attribute__((ext_vector_type(2)));
typedef float v8f __attribute__((ext_vector_type(8)));

#define NANTS  32
#define NBLS   64
#define NTIMES 16
#define NFREQS 32
#define NPIX   12288
#define NBEAM  4
// np.float32(2*pi / 299800000.0)
#define K_TWO_PI_OVER_C 2.0957923e-08f

// One block per (baseline, time). 256 threads, each owning 4 consecutive
// pixels per outer iteration (12288 = 12 * 1024). Frequency fringe advanced
// by complex rotation (uniform freq grid => phi(f) = phi0 + f*dphi), so only
// 2 sincos pairs per pixel instead of 32. Final 256-thread x 64-value block
// reduction is done on the matrix pipe with chained V_WMMA_F32_16X16X4_F32
// (ones as B => row sums of A).
__global__ __launch_bounds__(256) void rime_vis_kernel(
    const float* __restrict__ sky,        // [NFREQS][NPIX]
    const float* __restrict__ ant_beams,  // [NTIMES][1][1][NBEAM][NFREQS][NPIX]
    const float* __restrict__ topo,       // [NTIMES][3][NPIX]
    const float* __restrict__ antpos,     // [NANTS][3]
    const float* __restrict__ freqs,      // [NFREQS]
    const int*   __restrict__ bls,        // [NBLS][2]
    const int*   __restrict__ ant2beam,   // [NANTS]
    float*       __restrict__ out)        // [NBLS][NTIMES][NFREQS][2]
{
    __shared__ float red[256 * 64];   // 64 KB partials: [thread][f*2 + reim]

    const int blk = blockIdx.x;       // 0..1023
    const int bl  = blk >> 4;         // / NTIMES
    const int j   = blk & 15;         // % NTIMES
    const int tid = threadIdx.x;

    // Uniform per-block scalars (scalar loads)
    const int a0 = bls[2 * bl + 0];
    const int a1 = bls[2 * bl + 1];
    const float bx = antpos[3 * a1 + 0] - antpos[3 * a0 + 0];
    const float by = antpos[3 * a1 + 1] - antpos[3 * a0 + 1];
    const float bz = antpos[3 * a1 + 2] - antpos[3 * a0 + 2];
    const int beam1 = ant2beam[a0];
    const int beam2 = ant2beam[a1];
    const float kf0 = K_TWO_PI_OVER_C * freqs[0];
    const float kdf = K_TWO_PI_OVER_C * (freqs[1] - freqs[0]);

    const float* __restrict__ b1p = ant_beams + (size_t)(j * NBEAM + beam1) * (NFREQS * NPIX);
    const float* __restrict__ b2p = ant_beams + (size_t)(j * NBEAM + beam2) * (NFREQS * NPIX);
    const float* __restrict__ tp  = topo + (size_t)j * (3 * NPIX);

    float accRe[NFREQS], accIm[NFREQS];
#pragma unroll
    for (int f = 0; f < NFREQS; ++f) { accRe[f] = 0.0f; accIm[f] = 0.0f; }

    for (int it = 0; it < NPIX / 1024; ++it) {
        const int s0 = it * 1024 + tid * 4;

        const float4 tx = *(const float4*)(tp + 0 * NPIX + s0);
        const float4 ty = *(const float4*)(tp + 1 * NPIX + s0);
        const float4 tz = *(const float4*)(tp + 2 * NPIX + s0);

        float bd[4];
        bd[0] = bx * tx.x + by * ty.x + bz * tz.x;
        bd[1] = bx * tx.y + by * ty.y + bz * tz.y;
        bd[2] = bx * tx.z + by * ty.z + bz * tz.z;
        bd[3] = bx * tx.w + by * ty.w + bz * tz.w;

        float cr[4], ci[4], sr[4], si[4];
#pragma unroll
        for (int k = 0; k < 4; ++k) {
            __sincosf(kf0 * bd[k], &ci[k], &cr[k]);  // current fringe = e^{i*phi0}
            __sincosf(kdf * bd[k], &si[k], &sr[k]);  // step = e^{i*dphi}
        }

#pragma unroll
        for (int f = 0; f < NFREQS; ++f) {
            const float4 v1 = *(const float4*)(b1p + (size_t)f * NPIX + s0);
            const float4 v2 = *(const float4*)(b2p + (size_t)f * NPIX + s0);
            const float4 sk = *(const float4*)(sky + (size_t)f * NPIX + s0);

            float p[4];
            p[0] = v1.x * v2.x * sk.x;
            p[1] = v1.y * v2.y * sk.y;
            p[2] = v1.z * v2.z * sk.z;
            p[3] = v1.w * v2.w * sk.w;

            float re = accRe[f], im = accIm[f];
#pragma unroll
            for (int k = 0; k < 4; ++k) {
                re = fmaf(p[k], cr[k], re);
                im = fmaf(p[k], ci[k], im);
            }
            accRe[f] = re;
            accIm[f] = im;

            if (f != NFREQS - 1) {
#pragma unroll
                for (int k = 0; k < 4; ++k) {  // fringe *= step
                    const float ncr = cr[k] * sr[k] - ci[k] * si[k];
                    const float nci = cr[k] * si[k] + ci[k] * sr[k];
                    cr[k] = ncr;
                    ci[k] = nci;
                }
            }
        }
    }

    // Stage partials: red[t][2f] = Re, red[t][2f+1] = Im
#pragma unroll
    for (int f = 0; f < NFREQS; ++f) {
        red[tid * 64 + 2 * f + 0] = accRe[f];
        red[tid * 64 + 2 * f + 1] = accIm[f];
    }
    __syncthreads();

    // Wave 0: reduce the 64 columns over 256 threads on the matrix pipe.
    // A[m][k] = red[4c+k][g*16+m] (documented 16x4 f32 A layout), B = ones
    // => D[m][n] = sum_k A[m][k]; chain C over 64 K-chunks of 4 threads.
    if (tid < 32) {
        const int lane = tid;
        const int m    = lane & 15;
        const int koff = (lane >> 4) << 1;  // lanes 0-15 hold K=0,1; lanes 16-31 hold K=2,3

        v2f ones;
        ones.x = 1.0f;
        ones.y = 1.0f;

        for (int g = 0; g < 4; ++g) {       // 4 groups of 16 output values
            v8f acc = {};
            for (int c = 0; c < 64; ++c) {  // 64 chunks x 4 threads = 256 threads
                v2f a;
                a.x = red[(4 * c + koff + 0) * 64 + g * 16 + m];
                a.y = red[(4 * c + koff + 1) * 64 + g * 16 + m];
                acc = __builtin_amdgcn_wmma_f32_16x16x4_f32(
                    false, a, false, ones, (short)0, acc, false, false);
            }
            // C/D layout: VGPR i holds M=i (lanes 0-15, N=lane) and M=8+i
            // (lanes 16-31, N=lane-16). Read column N=0: lane 0 -> M=0..7,
            // lane 16 -> M=8..15.
            float* ob = out + (size_t)blk * 64 + g * 16;
            if (lane == 0) {
#pragma unroll
                for (int i = 0; i < 8; ++i) ob[i] = acc[i];
            } else if (lane == 16) {
#pragma unroll
                for (int i = 0; i < 8; ++i) ob[8 + i] = acc[i];
            }
        }
    }
}

extern "C" void kernel_launch(void* const* d_in, const int* in_sizes, int n_in,
                              void* d_out, int out_size, void* d_ws, size_t ws_size,
                              hipStream_t stream) {
    const float* sky       = (const float*)d_in[0];
    const float* ant_beams = (const float*)d_in[1];
    const float* topo      = (const float*)d_in[2];
    const float* antpos    = (const float*)d_in[3];
    const float* freqs     = (const float*)d_in[4];
    const int*   bls       = (const int*)d_in[5];
    const int*   ant2beam  = (const int*)d_in[6];
    float*       out       = (float*)d_out;

    rime_vis_kernel<<<dim3(NBLS * NTIMES), dim3(256), 0, stream>>>(
        sky, ant_beams, topo, antpos, freqs, bls, ant2beam, out);
}